// HyperComputeModule_11587821764851
// MI455X (gfx1250) — compile-verified
//
#include <hip/hip_runtime.h>
#include <hip/hip_bf16.h>
#include <stdint.h>
#include <math.h>

#define BB 4
#define CC 256
#define NN 4096          // H*W = 64*64
#define EE 64            // embed dim
#define KK 24
#define THRESH 10.0f
#define CHUNK 64
#define NCHUNK (NN / CHUNK)
#define ADJ_STRIDE 26    // up to 25 entries per row

typedef float v2f __attribute__((ext_vector_type(2)));
typedef float v8f __attribute__((ext_vector_type(8)));
typedef uint32_t u32x4 __attribute__((ext_vector_type(4)));
typedef int i32x4 __attribute__((ext_vector_type(4)));
typedef int i32x8 __attribute__((ext_vector_type(8)));

// LDS byte offset of a __shared__ object (low 32 bits of the LDS-aperture flat address)
__device__ __forceinline__ unsigned lds_off(const void* p) {
  return (unsigned)(uintptr_t)p;
}

// Issue a 2-D TDM tile load (Tensor Data Mover): global[row-major, stride0] -> LDS.
// Descriptor packing per CDNA5 ISA 8.3/8.4. data_size = 4 bytes. Optional LDS padding
// of 2 DWORDs after every 64 DWORDs (pad_interval code 5, pad_amount code 1) to give
// 66-float LDS row pitch for bank-conflict-free k-major reads.
__device__ __forceinline__ void tdm_load_2d(unsigned ldsAddr, const void* gptr,
                                            uint32_t td0, uint32_t td1,
                                            uint32_t tile0, uint32_t tile1,
                                            uint64_t stride0, int padded) {
  const uint64_t ga = (uint64_t)(uintptr_t)gptr;
  u32x4 g0;
  g0.x = 1u;                                                   // count=1, user mode
  g0.y = ldsAddr;                                              // lds_addr
  g0.z = (uint32_t)ga;                                         // global_addr[31:0]
  g0.w = (uint32_t)((ga >> 32) & 0x1FFFFFFull) | (2u << 30);   // global_addr[56:32] | type=2
  i32x8 g1;
  uint32_t w0 = (2u << 16);                                    // data_size = 4B, no multicast
  if (padded) w0 |= (1u << 20) | (5u << 22) | (1u << 25);      // pad 2 dwords / 64 dwords
  g1[0] = (int)w0;
  g1[1] = (int)((td0 & 0xFFFFu) << 16);                        // tensor_dim0 lo
  g1[2] = (int)(((td0 >> 16) & 0xFFFFu) | ((td1 & 0xFFFFu) << 16));
  g1[3] = (int)(((td1 >> 16) & 0xFFFFu) | ((tile0 & 0xFFFFu) << 16));
  g1[4] = (int)(tile1 & 0xFFFFu);                              // tile_dim1, tile_dim2=0
  g1[5] = (int)(uint32_t)(stride0 & 0xFFFFFFFFull);            // tensor_dim0_stride lo
  g1[6] = (int)(uint32_t)((stride0 >> 32) & 0xFFFFull);        // stride hi, dim1_stride=0
  g1[7] = 0;
  const i32x4 z4 = {0, 0, 0, 0};                               // groups 2/3 unused (2-D)
  const i32x8 z8 = {0, 0, 0, 0, 0, 0, 0, 0};
  __builtin_amdgcn_tensor_load_to_lds(g0, g1, z4, z4, z8, 0);  // 6-arg toolchain form
}

// ---------------------------------------------------------------- zero ws
__global__ __launch_bounds__(256) void k_zero(float* __restrict__ Y, int* __restrict__ DeI) {
  size_t i = (size_t)blockIdx.x * 256 + threadIdx.x;
  if (i < (size_t)BB * NN * CC) Y[i] = 0.0f;
  if (i < (size_t)BB * NN)      DeI[i] = 0;
}

// ---------------------------------------------------------------- fused front GEMMs (fp32 WMMA):
//   X_lin = x_flat @ fc_w^T + b   (4 column groups of 64)
//   emb   = x_flat @ dist_w^T     (1 column group of 64)
// A tile (16 x 256) staged once per block via TDM into [k][n] LDS layout.
__global__ __launch_bounds__(128)
void k_front(const float* __restrict__ x, const float* __restrict__ fcw,
             const float* __restrict__ fcb, const float* __restrict__ dw,
             float* __restrict__ Xlin, float* __restrict__ emb) {
  __shared__ float As2[CC][16];                 // 16 KB, [k][n]
  const int lane = threadIdx.x & 31, wave = threadIdx.x >> 5;
  const int l16 = lane & 15, half = lane >> 4;
  const int n0 = blockIdx.x * 16;
  const int b  = blockIdx.z;
  const float* xb = x + (size_t)b * CC * NN;

  if (threadIdx.x < 32) {                       // wave 0 drives the TDM
    tdm_load_2d(lds_off(&As2[0][0]), xb + n0, NN, CC, 16, CC, (uint64_t)NN, 0);
    __builtin_amdgcn_s_wait_tensorcnt(0);
  }
  __syncthreads();

  for (int g = 0; g < 5; ++g) {
    const int c0 = (g < 4) ? (g * 64 + wave * 16) : (wave * 16);
    const float* wrow = (g < 4) ? (fcw + (size_t)(c0 + l16) * CC)
                                : (dw  + (size_t)(c0 + l16) * CC);
    if (g < 4) {                                // prefetch next group's weights
      const float* nxt = (g + 1 < 4) ? (fcw + (size_t)((g + 1) * 64 + wave * 16 + l16) * CC)
                                     : (dw  + (size_t)(wave * 16 + l16) * CC);
      __builtin_prefetch(nxt, 0, 0);
    }
    v8f acc = {};
    for (int k0 = 0; k0 < CC; k0 += 4) {
      const int ka = k0 + 2 * half;
      v2f a, bmat;
      a.x = As2[ka + 0][l16];                   // A[n, k] fragment (bank-conflict-free)
      a.y = As2[ka + 1][l16];
      bmat.x = wrow[ka + 0];                    // B[k, c] = W[c, k]
      bmat.y = wrow[ka + 1];
      acc = __builtin_amdgcn_wmma_f32_16x16x4_f32(false, a, false, bmat, (short)0, acc, false, false);
    }
    if (g < 4) {
      const float bias = fcb[c0 + l16];
      float* orow = Xlin + (size_t)b * NN * CC;
#pragma unroll
      for (int r = 0; r < 8; ++r)
        orow[(size_t)(n0 + r + half * 8) * CC + c0 + l16] = acc[r] + bias;
    } else {
      float* orow = emb + (size_t)b * NN * EE;
#pragma unroll
      for (int r = 0; r < 8; ++r)
        orow[(size_t)(n0 + r + half * 8) * EE + c0 + l16] = acc[r];
    }
  }
}

// ---------------------------------------------------------------- squared norms
__global__ __launch_bounds__(256) void k_sq(const float* __restrict__ emb, float* __restrict__ sq) {
  const int i = blockIdx.x * 256 + threadIdx.x;   // over B*N
  const float* e = emb + (size_t)i * EE;
  float s = 0.0f;
#pragma unroll
  for (int k = 0; k < EE; ++k) s += e[k] * e[k];
  sq[i] = s;
}

// ---------------------------------------------------------------- Gram (WMMA) + streaming top-K
// Double-buffered TDM pipeline stages emb column chunks (64 cols x 64 k) into padded LDS.
__global__ __launch_bounds__(512)
void k_dist_topk(const float* __restrict__ embp, const float* __restrict__ sqp,
                 int* __restrict__ adjL, int* __restrict__ DvI, int* __restrict__ DeI) {
  __shared__ float embA[16][66];                // padded row pitch (TDM pad feature)
  __shared__ float embB[2][CHUNK][66];          // double-buffered column chunk
  __shared__ float distT[16][CHUNK];
  __shared__ float sqN[16];
  __shared__ float sqc[CHUNK];
  __shared__ float resVal[16][KK];
  __shared__ int   resIdx[16][KK];
  __shared__ int   cnt[16];
  __shared__ int   selfF[16];
  __shared__ int   rowList[16][32];

  const int tid = threadIdx.x;
  const int lane = tid & 31, wave = tid >> 5;   // 16 wave32s; wave w owns row w
  const int l16 = lane & 15, half = lane >> 4;
  const int n0 = blockIdx.x * 16;
  const int b  = blockIdx.y;
  const float* eb = embp + (size_t)b * NN * EE;

  if (wave == 0) {                              // prologue: A rows + chunk 0
    tdm_load_2d(lds_off(&embA[0][0]),    eb + (size_t)n0 * EE, EE, NN, EE, 16,    (uint64_t)EE, 1);
    tdm_load_2d(lds_off(&embB[0][0][0]), eb,                   EE, NN, EE, CHUNK, (uint64_t)EE, 1);
  }
  if (tid < 16) sqN[tid] = sqp[b * NN + n0 + tid];

  float bv[KK]; int bix[KK];
#pragma unroll
  for (int j = 0; j < KK; ++j) { bv[j] = 3.0e38f; bix[j] = -1; }
  float worstV = 3.0e38f; int worstJ = 0;

  for (int ci = 0; ci < NCHUNK; ++ci) {
    const int m0 = ci * CHUNK;
    if (wave == 0) {
      if (ci + 1 < NCHUNK) {                    // issue next chunk, then counted wait
        tdm_load_2d(lds_off(&embB[(ci + 1) & 1][0][0]), eb + (size_t)(m0 + CHUNK) * EE,
                    EE, NN, EE, CHUNK, (uint64_t)EE, 1);
        __builtin_amdgcn_s_wait_tensorcnt(1);   // chunk ci complete, ci+1 in flight
      } else {
        __builtin_amdgcn_s_wait_tensorcnt(0);
      }
    }
    if (tid < CHUNK) sqc[tid] = sqp[b * NN + m0 + tid];
    __syncthreads();

    // --- Gram tiles for this 16 x CHUNK slab (waves 0..3, one 16x16 tile each)
    if (wave < CHUNK / 16) {
      const int colb = wave * 16;
      const float (*Bt)[66] = embB[ci & 1];
      v8f acc = {};
      for (int k0 = 0; k0 < EE; k0 += 4) {
        const int ka = k0 + 2 * half;
        v2f a, bmat;
        a.x = embA[l16][ka + 0];
        a.y = embA[l16][ka + 1];
        bmat.x = Bt[colb + l16][ka + 0];        // B[k, m] = emb[m, k]
        bmat.y = Bt[colb + l16][ka + 1];
        acc = __builtin_amdgcn_wmma_f32_16x16x4_f32(false, a, false, bmat, (short)0, acc, false, false);
      }
      const float sm = sqc[colb + l16];
#pragma unroll
      for (int r = 0; r < 8; ++r) {
        const int row = r + half * 8;
        const float d2 = sqN[row] + sm - 2.0f * acc[r];
        distT[row][colb + l16] = sqrtf(fmaxf(d2, 0.0f));
      }
    }
    __syncthreads();

    // --- per-lane streaming top-24 (wave w scans row w)
    for (int j = 0; j < CHUNK / 32; ++j) {
      const int cidx = lane + j * 32;
      const float v = distT[wave][cidx];
      if (v < worstV) {
        bv[worstJ] = v; bix[worstJ] = m0 + cidx;
        worstV = -1.0f;
#pragma unroll
        for (int q = 0; q < KK; ++q) if (bv[q] > worstV) { worstV = bv[q]; worstJ = q; }
      }
    }
    __syncthreads();
  }

  // sort each lane's list ascending
#pragma unroll
  for (int i = 1; i < KK; ++i) {
    float v = bv[i]; int id = bix[i]; int j = i - 1;
    while (j >= 0 && bv[j] > v) { bv[j + 1] = bv[j]; bix[j + 1] = bix[j]; --j; }
    bv[j + 1] = v; bix[j + 1] = id;
  }
  // wave-wide 32-way merge, 24 rounds (repeated argmin over lane heads)
  int p = 0;
  for (int it = 0; it < KK; ++it) {
    float v = (p < KK) ? bv[p] : 3.0e38f;
    int id  = (p < KK) ? bix[p] : -1;
    int src = lane;
    for (int off = 16; off > 0; off >>= 1) {
      float ov = __shfl_xor(v, off, 32);
      int  oid = __shfl_xor(id, off, 32);
      int  os  = __shfl_xor(src, off, 32);
      if (ov < v || (ov == v && os < src)) { v = ov; id = oid; src = os; }
    }
    if (src == lane) ++p;
    if (lane == 0) { resVal[wave][it] = v; resIdx[wave][it] = id; }
  }
  if (lane == 0) { cnt[wave] = 0; selfF[wave] = 0; }
  __syncthreads();

  const int rowN = n0 + wave;
  if (lane < KK) {
    const float v = resVal[wave][lane]; const int id = resIdx[wave][lane];
    if (id >= 0 && v <= THRESH) {
      const int pos = atomicAdd(&cnt[wave], 1);
      rowList[wave][pos] = id;
      if (id == rowN) atomicAdd(&selfF[wave], 1);
    }
  }
  __syncthreads();
  if (lane == 0 && selfF[wave] == 0) { rowList[wave][cnt[wave]++] = rowN; }  // adj |= eye
  __syncthreads();
  const int cc = cnt[wave];
  if (lane == 0) DvI[b * NN + rowN] = cc;
  if (lane < cc) {
    const int e = rowList[wave][lane];
    adjL[((size_t)(b * NN + rowN)) * ADJ_STRIDE + lane] = e;
    atomicAdd(&DeI[b * NN + e], 1);
  }
}

// ---------------------------------------------------------------- Y = H^T @ X_lin (row-sparse scatter)
__global__ __launch_bounds__(256)
void k_scatter(const float* __restrict__ Xlin, const int* __restrict__ adjL,
               const int* __restrict__ DvI, float* __restrict__ Y) {
  const int bn = blockIdx.x;                // b*N + n
  const int c = threadIdx.x;
  const int b = bn / NN;
  const float xl = Xlin[(size_t)bn * CC + c];
  const int cc = DvI[bn];
  const int* lst = adjL + (size_t)bn * ADJ_STRIDE;
  for (int j = 0; j < cc; ++j) {
    const int e = lst[j];
    atomicAdd(&Y[((size_t)b * NN + e) * CC + c], xl);
  }
}

// ---------------------------------------------------------------- pre = H De^-1 Y / Dv + x_flat
__global__ __launch_bounds__(256)
void k_gather(const float* __restrict__ Y, const int* __restrict__ adjL,
              const int* __restrict__ DvI, const int* __restrict__ DeI,
              const float* __restrict__ x, float* __restrict__ pre) {
  const int bn = blockIdx.x;
  const int c = threadIdx.x;
  const int b = bn / NN, n = bn % NN;
  const int cc = DvI[bn];
  const int* lst = adjL + (size_t)bn * ADJ_STRIDE;
  float acc = 0.0f;
  for (int j = 0; j < cc; ++j) {
    const int e = lst[j];
    const float de = fmaxf((float)DeI[b * NN + e], 1.0f);
    acc += Y[((size_t)b * NN + e) * CC + c] / de;
  }
  const float dv = fmaxf((float)cc, 1.0f);
  pre[(size_t)bn * CC + c] = acc / dv + x[((size_t)b * CC + c) * NN + n];
}

// ---------------------------------------------------------------- BN batch stats per channel
__global__ __launch_bounds__(256)
void k_stats(const float* __restrict__ pre, float* __restrict__ meanA, float* __restrict__ rsigA) {
  const int c = blockIdx.x, t = threadIdx.x;
  float s = 0.0f, s2 = 0.0f;
  for (int i = t; i < BB * NN; i += 256) {
    const float v = pre[(size_t)i * CC + c];
    s += v; s2 += v * v;
  }
  __shared__ float S[256], S2[256];
  S[t] = s; S2[t] = s2; __syncthreads();
  for (int o = 128; o > 0; o >>= 1) {
    if (t < o) { S[t] += S[t + o]; S2[t] += S2[t + o]; }
    __syncthreads();
  }
  if (t == 0) {
    const float inv = 1.0f / (float)(BB * NN);
    const float m = S[0] * inv;
    const float var = S2[0] * inv - m * m;
    meanA[c] = m;
    rsigA[c] = rsqrtf(var + 1e-5f);
  }
}

// ---------------------------------------------------------------- BN apply + SiLU + transpose out
__global__ __launch_bounds__(256)
void k_out(const float* __restrict__ pre, const float* __restrict__ meanA,
           const float* __restrict__ rsigA, const float* __restrict__ g,
           const float* __restrict__ be, float* __restrict__ out) {
  const size_t i = (size_t)blockIdx.x * 256 + threadIdx.x;  // over B*C*N, n fastest
  if (i >= (size_t)BB * CC * NN) return;
  const int n = (int)(i % NN);
  const size_t bc = i / NN;
  const int c = (int)(bc % CC), b = (int)(bc / CC);
  const float v = pre[((size_t)b * NN + n) * CC + c];
  const float o = (v - meanA[c]) * rsigA[c] * g[c] + be[c];
  out[i] = o / (1.0f + expf(-o));   // SiLU
}

// ----------------------------------------------------------------
extern "C" void kernel_launch(void* const* d_in, const int* in_sizes, int n_in,
                              void* d_out, int out_size, void* d_ws, size_t ws_size,
                              hipStream_t stream) {
  const float* x    = (const float*)d_in[0];
  const float* fcw  = (const float*)d_in[1];
  const float* fcb  = (const float*)d_in[2];
  const float* dw   = (const float*)d_in[3];
  const float* gam  = (const float*)d_in[4];
  const float* bet  = (const float*)d_in[5];
  float* out = (float*)d_out;

  // workspace carve-up (~56 MB)
  float* Xlin = (float*)d_ws;                       // B*N*C
  float* emb  = Xlin + (size_t)BB * NN * CC;        // B*N*E
  float* sq   = emb  + (size_t)BB * NN * EE;        // B*N
  float* Y    = sq   + (size_t)BB * NN;             // B*N*C
  float* pre  = Y    + (size_t)BB * NN * CC;        // B*N*C
  float* meanA = pre + (size_t)BB * NN * CC;        // C
  float* rsigA = meanA + CC;                        // C
  int* adjL = (int*)(rsigA + CC);                   // B*N*ADJ_STRIDE
  int* DvI  = adjL + (size_t)BB * NN * ADJ_STRIDE;  // B*N
  int* DeI  = DvI + (size_t)BB * NN;                // B*N

  k_zero<<<(BB * NN * CC + 255) / 256, 256, 0, stream>>>(Y, DeI);
  k_front<<<dim3(NN / 16, 1, BB), 128, 0, stream>>>(x, fcw, fcb, dw, Xlin, emb);
  k_sq  <<<(BB * NN) / 256, 256, 0, stream>>>(emb, sq);
  k_dist_topk<<<dim3(NN / 16, BB), 512, 0, stream>>>(emb, sq, adjL, DvI, DeI);
  k_scatter<<<BB * NN, 256, 0, stream>>>(Xlin, adjL, DvI, Y);
  k_gather <<<BB * NN, 256, 0, stream>>>(Y, adjL, DvI, DeI, x, pre);
  k_stats  <<<CC, 256, 0, stream>>>(pre, meanA, rsigA);
  k_out    <<<(BB * CC * NN + 255) / 256, 256, 0, stream>>>(pre, meanA, rsigA, gam, bet, out);
}